// GAT_Layer_54700703481984
// MI455X (gfx1250) — compile-verified
//
#include <hip/hip_runtime.h>
#include <math.h>

// ---------------------------------------------------------------------------
// GAT layer for MI455X (gfx1250, wave32).
//  x: (N,256) f32   edge_index: (2,E) int   W: (256,256) f32   b: (256,) f32
//  a_src/a_trg: (8,32) f32
//  out: (N,256) f32
//
// Pipeline:
//  K0 init      : zero out / denom / maxkey
//  K1 gemm      : h_prime = x @ W + b        (V_WMMA_F32_16X16X4_F32,
//                 64-row A panel staged in LDS via global_load_async_to_lds_b128,
//                 4 M-tiles per block so every B fragment feeds 4 WMMAs)
//  K2 attnproj  : h_src/h_trg = <h_prime, a_src/a_trg> per (node,head)
//  K3 edge_e    : e = leaky_relu(h_src[s]+h_trg[t]); global max (atomicMax key)
//  K4 expdenom  : exp_e = exp(e - gmax); denom[t,k] += exp_e  (atomic f32)
//  K5 scatter   : out[t,:] += h_prime[s,:] * alpha            (atomic f32)
// ---------------------------------------------------------------------------

typedef __attribute__((ext_vector_type(2))) float v2f;
typedef __attribute__((ext_vector_type(4))) float v4f;
typedef __attribute__((ext_vector_type(8))) float v8f;

#define FIN   256
#define KOUT  256   // K * F_OUT
#define NHEAD 8
#define FOUT  32
#define NEG_SLOPE 0.2f
#define MT    4     // M-tiles (16 rows each) per block -> 64-row panel

// ---- order-preserving float <-> uint key (works for negatives) -------------
__device__ __forceinline__ unsigned fkey(float f) {
  unsigned u = __float_as_uint(f);
  return (u & 0x80000000u) ? ~u : (u | 0x80000000u);
}
__device__ __forceinline__ float funkey(unsigned u) {
  return (u & 0x80000000u) ? __uint_as_float(u & 0x7fffffffu)
                           : __uint_as_float(~u);
}

// ---- K0: init --------------------------------------------------------------
__global__ void gat_init_kernel(float* __restrict__ out, float* __restrict__ denom,
                                unsigned* __restrict__ maxkey, int n_out, int n_denom) {
  int i = blockIdx.x * blockDim.x + threadIdx.x;
  if (i < n_out)   out[i]   = 0.0f;
  if (i < n_denom) denom[i] = 0.0f;
  if (i == 0)      *maxkey  = 0u;   // unsigned-min; any real key exceeds this
}

// ---- K1: GEMM via V_WMMA_F32_16X16X4_F32 -----------------------------------
// Block = 64-row A panel (4 M-tiles) x all 256 N columns.
// 8 waves x 2 N-tiles each; every wave keeps 4x2 accumulators so each loaded
// B fragment is consumed by 4 WMMAs (4x less global B traffic than 1 M-tile).
//
// f32 A 16x4 layout : lanes 0-15 -> M=lane,   VGPRs hold K={0,1}
//                     lanes 16-31-> M=lane-16,VGPRs hold K={2,3}
// f32 B 4x16 layout : mirrored (lane half selects K pair, lane&15 = N col)
// f32 C/D 16x16     : VGPR v -> M = v + (lane>=16 ? 8:0), N = lane&15
__global__ __launch_bounds__(256) void gat_gemm_kernel(
    const float* __restrict__ x, const float* __restrict__ W,
    const float* __restrict__ bias, float* __restrict__ h_prime, int Nn) {
  __shared__ float xtile[MT * 16 * FIN];     // 64 KB A panel

  const int tid  = threadIdx.x;
  const int wave = tid >> 5;                 // 0..7
  const int lane = tid & 31;
  const int m0   = blockIdx.x * (MT * 16);
  const int col  = lane & 15;
  const int khalf = (lane >> 4) << 1;        // 0 or 2

  // ---- stage A panel: 64KB contiguous (rows m0..m0+63 of x) ----------------
  {
    const size_t lim = (size_t)Nn * FIN * 4;           // total bytes of x
#if defined(__gfx1250__)
    const unsigned lbase = (unsigned)(uintptr_t)&xtile[0];  // LDS byte offset
#pragma unroll
    for (int p = 0; p < MT * 4; ++p) {
      size_t go = (size_t)m0 * FIN * 4 + (size_t)tid * 16 + (size_t)p * 4096;
      if (go + 16 > lim) go = lim - 16;                // clamp (last block)
      unsigned lofs  = lbase + (unsigned)(tid * 16 + p * 4096);
      unsigned long long gaddr = (unsigned long long)((const char*)x + go);
      asm volatile("global_load_async_to_lds_b128 %0, %1, off"
                   :: "v"(lofs), "v"(gaddr) : "memory");
    }
    asm volatile("s_wait_asynccnt 0x0" ::: "memory");
#else
    // host-pass / non-gfx1250 fallback: plain copy
    for (int p = 0; p < MT * 4; ++p) {
      size_t go = (size_t)m0 * FIN * 4 + (size_t)tid * 16 + (size_t)p * 4096;
      if (go + 16 > lim) go = lim - 16;
      v4f v = *(const v4f*)((const char*)x + go);
      *(v4f*)((char*)xtile + tid * 16 + p * 4096) = v;
    }
#endif
    __syncthreads();
  }

  const int n0 = (wave * 2) * 16;
  const int n1 = n0 + 16;

  v8f acc[MT][2];
#pragma unroll
  for (int mt = 0; mt < MT; ++mt) {
    acc[mt][0] = (v8f){0.f,0.f,0.f,0.f,0.f,0.f,0.f,0.f};
    acc[mt][1] = (v8f){0.f,0.f,0.f,0.f,0.f,0.f,0.f,0.f};
  }

#pragma unroll 4
  for (int k0 = 0; k0 < FIN; k0 += 4) {
    // B fragments (global, L2-resident W), shared by all 4 M-tiles
    const float* __restrict__ wr0 = W + (size_t)(k0 + khalf) * KOUT;
    const float* __restrict__ wr1 = wr0 + KOUT;
    v2f b0, b1;
    b0.x = wr0[n0 + col];
    b0.y = wr1[n0 + col];
    b1.x = wr0[n1 + col];
    b1.y = wr1[n1 + col];

    // A fragments (LDS), one per M-tile
    v2f a[MT];
#pragma unroll
    for (int mt = 0; mt < MT; ++mt) {
      const float* __restrict__ arow = &xtile[((size_t)mt * 16 + col) * FIN];
      a[mt].x = arow[k0 + khalf];            // ds_load_b64
      a[mt].y = arow[k0 + khalf + 1];
    }

#pragma unroll
    for (int mt = 0; mt < MT; ++mt) {
      acc[mt][0] = __builtin_amdgcn_wmma_f32_16x16x4_f32(
          false, a[mt], false, b0, (short)0, acc[mt][0], false, false);
      acc[mt][1] = __builtin_amdgcn_wmma_f32_16x16x4_f32(
          false, a[mt], false, b1, (short)0, acc[mt][1], false, false);
    }
  }

  const float bv0 = bias[n0 + col];
  const float bv1 = bias[n1 + col];
  const int rsub = (lane >> 4) << 3;           // +0 or +8
#pragma unroll
  for (int mt = 0; mt < MT; ++mt) {
    const int rbase = m0 + mt * 16 + rsub;
#pragma unroll
    for (int v = 0; v < 8; ++v) {
      int r = rbase + v;
      if (r < Nn) {
        h_prime[(size_t)r * KOUT + n0 + col] = acc[mt][0][v] + bv0;
        h_prime[(size_t)r * KOUT + n1 + col] = acc[mt][1][v] + bv1;
      }
    }
  }
}

// ---- K2: per-(node,head) attention projections -----------------------------
__global__ void gat_attnproj_kernel(const float* __restrict__ h_prime,
                                    const float* __restrict__ a_src,
                                    const float* __restrict__ a_trg,
                                    float* __restrict__ h_src,
                                    float* __restrict__ h_trg, int NK) {
  int idx = blockIdx.x * blockDim.x + threadIdx.x;   // n*8 + k
  if (idx >= NK) return;
  const int k = idx & (NHEAD - 1);
  const float* __restrict__ hp = h_prime + (size_t)idx * FOUT;
  const float* __restrict__ as = a_src + k * FOUT;
  const float* __restrict__ at = a_trg + k * FOUT;
  float s = 0.f, t = 0.f;
#pragma unroll
  for (int f = 0; f < FOUT; ++f) {
    float h = hp[f];
    s += h * as[f];
    t += h * at[f];
  }
  h_src[idx] = s;
  h_trg[idx] = t;
}

// ---- K3: edge scores + global max ------------------------------------------
__global__ __launch_bounds__(256) void gat_edge_e_kernel(
    const int* __restrict__ src, const int* __restrict__ trg, int E_, int total,
    const float* __restrict__ h_src, const float* __restrict__ h_trg,
    float* __restrict__ e_buf, unsigned* __restrict__ maxkey) {
  __shared__ unsigned smax[256];
  int i = blockIdx.x * blockDim.x + threadIdx.x;
  float lmax = -3.402823466e38f;
  if (i < total) {
    int s = (i < E_) ? src[i] : (i - E_);
    int t = (i < E_) ? trg[i] : (i - E_);
    const float* __restrict__ hs = h_src + (size_t)s * NHEAD;
    const float* __restrict__ ht = h_trg + (size_t)t * NHEAD;
#pragma unroll
    for (int k = 0; k < NHEAD; ++k) {
      float e = hs[k] + ht[k];
      e = (e > 0.f) ? e : NEG_SLOPE * e;         // leaky relu
      e_buf[(size_t)i * NHEAD + k] = e;
      lmax = fmaxf(lmax, e);
    }
  }
  smax[threadIdx.x] = fkey(lmax);
  __syncthreads();
  for (int off = 128; off > 0; off >>= 1) {
    if ((int)threadIdx.x < off) {
      unsigned o = smax[threadIdx.x + off];
      if (o > smax[threadIdx.x]) smax[threadIdx.x] = o;
    }
    __syncthreads();
  }
  if (threadIdx.x == 0) atomicMax(maxkey, smax[0]);
}

// ---- K4: exp + segment-sum denominator -------------------------------------
__global__ void gat_expdenom_kernel(const int* __restrict__ trg, int E_, int total,
                                    const unsigned* __restrict__ maxkey,
                                    float* __restrict__ e_buf,
                                    float* __restrict__ denom) {
  int i = blockIdx.x * blockDim.x + threadIdx.x;
  if (i >= total) return;
  const float gmax = funkey(*maxkey);
  int t = (i < E_) ? trg[i] : (i - E_);
#pragma unroll
  for (int k = 0; k < NHEAD; ++k) {
    float v = __expf(e_buf[(size_t)i * NHEAD + k] - gmax);
    e_buf[(size_t)i * NHEAD + k] = v;
    atomicAdd(&denom[(size_t)t * NHEAD + k], v);
  }
}

// ---- K5: weighted message scatter ------------------------------------------
// One block per (edge-or-loop), 256 threads = 8 heads x 32 features.
__global__ __launch_bounds__(256) void gat_scatter_kernel(
    const int* __restrict__ src, const int* __restrict__ trg, int E_, int total,
    const float* __restrict__ h_prime, const float* __restrict__ e_buf,
    const float* __restrict__ denom, float* __restrict__ out) {
  int i = blockIdx.x;
  if (i >= total) return;
  int s = (i < E_) ? src[i] : (i - E_);
  int t = (i < E_) ? trg[i] : (i - E_);
  int j = threadIdx.x;                 // feature index in [0,256)
  int k = j >> 5;                      // head
  float alpha = e_buf[(size_t)i * NHEAD + k] /
                (denom[(size_t)t * NHEAD + k] + 1e-10f);
  float m = h_prime[(size_t)s * KOUT + j] * alpha;
  atomicAdd(&out[(size_t)t * KOUT + j], m);
}

// ---------------------------------------------------------------------------
extern "C" void kernel_launch(void* const* d_in, const int* in_sizes, int n_in,
                              void* d_out, int out_size, void* d_ws, size_t ws_size,
                              hipStream_t stream) {
  const float* x     = (const float*)d_in[0];
  const int*   eidx  = (const int*)d_in[1];      // (2,E) row-major
  const float* W     = (const float*)d_in[2];
  const float* bias  = (const float*)d_in[3];
  const float* a_src = (const float*)d_in[4];
  const float* a_trg = (const float*)d_in[5];
  float* out = (float*)d_out;

  const int Nn    = in_sizes[0] / FIN;
  const int E_    = in_sizes[1] / 2;
  const int total = E_ + Nn;
  const int* src = eidx;
  const int* trg = eidx + E_;

  // Workspace partition (256B aligned)
  auto align256 = [](size_t v) { return (v + 255) & ~(size_t)255; };
  char* ws = (char*)d_ws;
  size_t off = 0;
  float* h_prime = (float*)(ws + off); off = align256(off + (size_t)Nn * KOUT * 4);
  float* h_src   = (float*)(ws + off); off = align256(off + (size_t)Nn * NHEAD * 4);
  float* h_trg   = (float*)(ws + off); off = align256(off + (size_t)Nn * NHEAD * 4);
  float* denom   = (float*)(ws + off); off = align256(off + (size_t)Nn * NHEAD * 4);
  float* e_buf   = (float*)(ws + off); off = align256(off + (size_t)total * NHEAD * 4);
  unsigned* maxkey = (unsigned*)(ws + off); off = align256(off + 4);
  (void)ws_size;

  // K0: init out / denom / maxkey
  {
    int n = Nn * KOUT;
    gat_init_kernel<<<(n + 255) / 256, 256, 0, stream>>>(out, denom, maxkey, n,
                                                         Nn * NHEAD);
  }
  // K1: WMMA GEMM (async-LDS A panel, 4 M-tiles/block)
  {
    int mblocks = (Nn + MT * 16 - 1) / (MT * 16);
    gat_gemm_kernel<<<mblocks, 256, 0, stream>>>(x, W, bias, h_prime, Nn);
  }
  // K2: attention projections
  {
    int nk = Nn * NHEAD;
    gat_attnproj_kernel<<<(nk + 255) / 256, 256, 0, stream>>>(
        h_prime, a_src, a_trg, h_src, h_trg, nk);
  }
  // K3: edge scores + global max
  gat_edge_e_kernel<<<(total + 255) / 256, 256, 0, stream>>>(
      src, trg, E_, total, h_src, h_trg, e_buf, maxkey);
  // K4: exp + denom
  gat_expdenom_kernel<<<(total + 255) / 256, 256, 0, stream>>>(
      trg, E_, total, maxkey, e_buf, denom);
  // K5: scatter messages
  gat_scatter_kernel<<<total, 256, 0, stream>>>(
      src, trg, E_, total, h_prime, e_buf, denom, out);
}